// PermutationMatrix_584115552806
// MI455X (gfx1250) — compile-verified
//
#include <hip/hip_runtime.h>
#include <hip/hip_bf16.h>

typedef __attribute__((ext_vector_type(2))) float v2f;
typedef __attribute__((ext_vector_type(8))) float v8f;

#define N 512
#define NN (N * N)
#define TEMP 50.0f
#define ITERS 1000
#define SK_BLOCKS 64  // persistent-kernel grid; one wave per matrix row (64*8 waves)

// ---------------------------------------------------------------------------
// Agent-scope (device-coherent) scalar access helpers. These lower to
// SCOPE=DEV vector memory ops that bypass L0, so shared vectors u/v are
// always L2-fresh WITHOUT device-wide fences -- leaving the read-only m0
// tiles resident in the WGP cache across all 2000 phases.
// ---------------------------------------------------------------------------
__device__ __forceinline__ float agent_load(const float* p) {
  return __hip_atomic_load(p, __ATOMIC_RELAXED, __HIP_MEMORY_SCOPE_AGENT);
}
__device__ __forceinline__ void agent_store(float* p, float x) {
  __hip_atomic_store(p, x, __ATOMIC_RELAXED, __HIP_MEMORY_SCOPE_AGENT);
}

// ---------------------------------------------------------------------------
// Grid-wide barrier (agent scope). Monotonic counter; each call advances the
// per-thread target by gridDim. Release add / acquire spin by thread 0 only;
// s_sleep in the spin to be polite to the SIMD. __syncthreads() before the
// arrive waits each wave's DEV-scope stores to the L2 coherence point.
// ---------------------------------------------------------------------------
__device__ __forceinline__ void gsync(unsigned* cnt, unsigned& target, unsigned nblk) {
  __syncthreads();
  target += nblk;
  if (threadIdx.x == 0) {
    __hip_atomic_fetch_add(cnt, 1u, __ATOMIC_RELEASE, __HIP_MEMORY_SCOPE_AGENT);
    while (__hip_atomic_load(cnt, __ATOMIC_ACQUIRE, __HIP_MEMORY_SCOPE_AGENT) < target) {
      __builtin_amdgcn_s_sleep(1);
    }
  }
  __syncthreads();
}

// ---------------------------------------------------------------------------
// Kernel 1: global max of M (single block, 1024 threads) + state init.
// ---------------------------------------------------------------------------
__global__ void k_max_init(const float* __restrict__ M, float* __restrict__ mx,
                           float* __restrict__ v, unsigned* __restrict__ cnt) {
  __shared__ float red[1024];
  int t = threadIdx.x;
  float m = -3.402823466e38f;
  for (int i = t; i < NN; i += 1024) m = fmaxf(m, M[i]);
  red[t] = m;
  __syncthreads();
  for (int s = 512; s > 0; s >>= 1) {
    if (t < s) red[t] = fmaxf(red[t], red[t + s]);
    __syncthreads();
  }
  if (t == 0) {
    mx[0] = red[0];
    cnt[0] = 0u;
  }
  if (t < N) v[t] = 1.0f;
}

// ---------------------------------------------------------------------------
// Kernel 2: m0 = exp(T*(M - max)), plus transposed copy for the col phase.
// ---------------------------------------------------------------------------
__global__ void k_exp(const float* __restrict__ M, const float* __restrict__ mx,
                      float* __restrict__ m0, float* __restrict__ m0T) {
  int idx = blockIdx.x * blockDim.x + threadIdx.x;
  float e = expf(TEMP * (M[idx] - mx[0]));
  int i = idx >> 9;
  int j = idx & (N - 1);
  m0[idx] = e;
  m0T[j * N + i] = e;
}

// ---------------------------------------------------------------------------
// Kernel 3: persistent Sinkhorn in scaling-vector form.
//   u = 1/(m0 @ v) ; v = 1/(m0^T @ u), 1000 iterations, grid barrier between
// phases. 64 blocks * 8 waves = 512 waves -> one wave owns one row.
// m0 rows (16 KB/block) stay hot in WGP$; only u/v (2 KB) cross L2.
// ---------------------------------------------------------------------------
__global__ void k_sinkhorn(const float* __restrict__ m0, const float* __restrict__ m0T,
                           float* __restrict__ u, float* __restrict__ v,
                           unsigned* __restrict__ cnt) {
  __shared__ float vec[N];
  const unsigned nblk = SK_BLOCKS;
  unsigned target = 0;
  int wave = threadIdx.x >> 5;
  int lane = threadIdx.x & 31;
  int row = blockIdx.x * 8 + wave;
  const float* __restrict__ r0 = m0 + row * N;
  const float* __restrict__ r1 = m0T + row * N;

  for (int it = 0; it < ITERS; ++it) {
    // ---- phase A: u[row] = 1 / dot(m0[row,:], v) ----
    for (int i = threadIdx.x; i < N; i += 256) vec[i] = agent_load(v + i);
    __syncthreads();
    float acc = 0.0f;
#pragma unroll
    for (int c = 0; c < 4; ++c) {
      int col = c * 128 + lane * 4;
      float4 mv = *(const float4*)(r0 + col);
      acc += mv.x * vec[col] + mv.y * vec[col + 1] + mv.z * vec[col + 2] + mv.w * vec[col + 3];
    }
#pragma unroll
    for (int off = 16; off > 0; off >>= 1) acc += __shfl_down(acc, off, 32);
    if (lane == 0) agent_store(u + row, 1.0f / acc);
    gsync(cnt, target, nblk);

    // ---- phase B: v[row] = 1 / dot(m0T[row,:], u) ----
    for (int i = threadIdx.x; i < N; i += 256) vec[i] = agent_load(u + i);
    __syncthreads();
    acc = 0.0f;
#pragma unroll
    for (int c = 0; c < 4; ++c) {
      int col = c * 128 + lane * 4;
      float4 mv = *(const float4*)(r1 + col);
      acc += mv.x * vec[col] + mv.y * vec[col + 1] + mv.z * vec[col + 2] + mv.w * vec[col + 3];
    }
#pragma unroll
    for (int off = 16; off > 0; off >>= 1) acc += __shfl_down(acc, off, 32);
    if (lane == 0) agent_store(v + row, 1.0f / acc);
    gsync(cnt, target, nblk);
  }
}

// ---------------------------------------------------------------------------
// Kernel 4: P = diag(u) m0 diag(v); T1 = P @ tril(ones) = reverse row cumsum.
// One thread per row (512 threads).
// ---------------------------------------------------------------------------
__global__ void k_finalize(const float* __restrict__ m0, const float* __restrict__ u,
                           const float* __restrict__ v, float* __restrict__ P,
                           float* __restrict__ T1) {
  int i = blockIdx.x * blockDim.x + threadIdx.x;
  if (i >= N) return;
  float ui = u[i];
  float s = 0.0f;
  const float* mrow = m0 + i * N;
  float* prow = P + i * N;
  float* trow = T1 + i * N;
  for (int c = N - 1; c >= 0; --c) {
    float p = ui * mrow[c] * v[c];
    prow[c] = p;
    s += p;
    trow[c] = s;  // T1[i][c] = sum_{m>=c} P[i][m]
  }
}

// ---------------------------------------------------------------------------
// Kernel 5: out = P @ T1^T via V_WMMA_F32_16X16X4_F32.
// out[i][j] = sum_k P[i][k] * T1[j][k]  ( == ((P L) P^T)^T [i][j] )
// One wave computes one 16x16 tile; 32x32 tiles total.
// A(16x4) lane layout: m = lane&15, kpair = (lane>>4)*2  -> {A[m][k+kp], A[m][k+kp+1]}
// B(4x16) lane layout: n = lane&15, kpair = (lane>>4)*2  -> {B[k+kp][n], ...} = T1[n][k+kp]
// D(16x16): vgpr r -> row = r + 8*(lane>>4), col = lane&15.
// ---------------------------------------------------------------------------
__global__ void k_gemm(const float* __restrict__ P, const float* __restrict__ T1,
                       float* __restrict__ out) {
  int gwave = (blockIdx.x * blockDim.x + threadIdx.x) >> 5;  // 0..1023
  int lane = threadIdx.x & 31;
  int tr = gwave >> 5;  // tile row 0..31
  int tc = gwave & 31;  // tile col 0..31
  int row0 = tr * 16;
  int col0 = tc * 16;
  int m = lane & 15;
  int kh = lane >> 4;  // 0 or 1

  const float* __restrict__ Arow = P + (row0 + m) * N;   // A-operand source row
  const float* __restrict__ Brow = T1 + (col0 + m) * N;  // B-operand: T1^T column = T1 row

  v8f acc = {0.f, 0.f, 0.f, 0.f, 0.f, 0.f, 0.f, 0.f};
#pragma unroll 4
  for (int k = 0; k < N; k += 4) {
    int ka = k + 2 * kh;
    v2f a = *(const v2f*)(Arow + ka);
    v2f b = *(const v2f*)(Brow + ka);
    acc = __builtin_amdgcn_wmma_f32_16x16x4_f32(false, a, false, b, (short)0, acc,
                                                false, false);
  }
#pragma unroll
  for (int r = 0; r < 8; ++r) {
    int orow = row0 + r + 8 * kh;
    out[orow * N + col0 + m] = acc[r];
  }
}

// ---------------------------------------------------------------------------
extern "C" void kernel_launch(void* const* d_in, const int* in_sizes, int n_in,
                              void* d_out, int out_size, void* d_ws, size_t ws_size,
                              hipStream_t stream) {
  const float* M = (const float*)d_in[0];
  // d_in[1] (lower) is tril(ones): folded analytically into the reverse cumsum.
  float* ws = (float*)d_ws;
  float* m0 = ws;            // [NN]
  float* m0T = ws + NN;      // [NN]
  float* P = ws + 2 * NN;    // [NN]
  float* T1 = ws + 3 * NN;   // [NN]
  float* u = ws + 4 * NN;    // [N]
  float* v = u + N;          // [N]
  float* mx = v + N;         // [1]
  unsigned* cnt = (unsigned*)(mx + 1);
  float* out = (float*)d_out;

  k_max_init<<<1, 1024, 0, stream>>>(M, mx, v, cnt);
  k_exp<<<NN / 256, 256, 0, stream>>>(M, mx, m0, m0T);
  k_sinkhorn<<<SK_BLOCKS, 256, 0, stream>>>(m0, m0T, u, v, cnt);
  k_finalize<<<2, 256, 0, stream>>>(m0, u, v, P, T1);
  k_gemm<<<128, 256, 0, stream>>>(P, T1, out);
}